// NewtonNet_6339371729510
// MI455X (gfx1250) — compile-verified
//
#include <hip/hip_runtime.h>
#include <math.h>

#define BB    8
#define AA    256
#define NNB   64
#define NFD   128
#define RESD  20
#define NHEAD 4
#define CUT_R 5.0f
#define PI_F  3.14159265358979323846f

typedef float v2f __attribute__((ext_vector_type(2)));
typedef float v8f __attribute__((ext_vector_type(8)));

// Global (address_space(1)) pointer helpers: force global_load/global_store
// instead of flat_load/flat_store (FLAT ties up DScnt + aperture checks).
#define AS1 __attribute__((address_space(1)))
typedef const AS1 float* gcf;
typedef       AS1 float* gf;
typedef const AS1 int*   gci;
__device__ __forceinline__ gcf togc(const float* p) { return (gcf)p; }
__device__ __forceinline__ gf  tog (float* p)       { return (gf)p; }
__device__ __forceinline__ gci togi(const int* p)   { return (gci)p; }

// ---------------------------------------------------------------------------
// Fused GEMM stage: OUT = act(X @ W + bias).
// Each wave computes a 64x32 output tile as 4x2 WMMA 16x16 subtiles,
// V_WMMA_F32_16X16X4_F32 over K. A fragments reused across 2 col-tiles,
// B fragments across 4 row-tiles -> 8 VMEM per 8 WMMAs, and with NC/K as
// compile-time constants the strides fold into load-offset immediates.
//  MODE 0: X row r = X0 + r*K                              (plain)
//  MODE 1: X row r = a_msij[b, N[r], :]  (gathered, K=128) (k-MLP input)
//  MODE 2: cols 0..127 as MODE 1, cols 128.. from X1 row r (v-MLP concat)
// ACT: 0 none, 1 relu, 2 silu
// ---------------------------------------------------------------------------
template<int MODE, int K, int NC, int ACT>
__global__ __launch_bounds__(256)
void wmma_mlp_gemm(const float* __restrict__ pX0,
                   const float* __restrict__ pX1,
                   const int*   __restrict__ pN,
                   const float* __restrict__ pW,
                   const float* __restrict__ pBias,
                   float* __restrict__ pOut,
                   int R)
{
    gcf X0 = togc(pX0);
    gcf X1 = togc(pX1);
    gci NI = togi(pN);
    gcf W  = togc(pW);
    gf  OUT = tog(pOut);
    const bool hasBias = (pBias != nullptr);
    gcf BIAS = togc(pBias);

    const int  lane  = threadIdx.x & 31;
    const int  wave  = threadIdx.x >> 5;
    constexpr int nBlk = NC >> 5;                     // 32-col blocks (pow2)
    const long tile  = (long)blockIdx.x * 8 + wave;
    const long total = ((long)R >> 6) * nBlk;         // 64-row x 32-col tiles
    if (tile >= total) return;  // wave-uniform; active waves keep EXEC all-1s

    const int  nB    = (int)(tile & (nBlk - 1)) * 32;
    const long mB    = (tile >> (nBlk == 4 ? 2 : 3)) * 64;
    const int  nl    = lane & 15;
    const int  khalf = (lane >> 4) << 1;              // lanes 16..31: K+2,K+3

    // ---- B operand: running column pointers + bias ----
    const int col0 = nB + nl;
    const int col1 = col0 + 16;
    gcf bp0 = W + (long)khalf * NC + col0;
    gcf bp1 = bp0 + 16;
    const float bv0 = hasBias ? BIAS[col0] : 0.0f;
    const float bv1 = hasBias ? BIAS[col1] : 0.0f;

    // ---- A operand: per-subtile row pointers (+ second segment for MODE 2) --
    gcf a0, a1, a2, a3;
    gcf c0 = nullptr, c1 = nullptr, c2 = nullptr, c3 = nullptr;
    {
        const long r0 = mB + 0 * 16 + nl;
        const long r1 = mB + 1 * 16 + nl;
        const long r2 = mB + 2 * 16 + nl;
        const long r3 = mB + 3 * 16 + nl;
        if (MODE == 0) {
            a0 = X0 + r0 * (long)K + khalf;
            a1 = X0 + r1 * (long)K + khalf;
            a2 = X0 + r2 * (long)K + khalf;
            a3 = X0 + r3 * (long)K + khalf;
        } else {
            const int b = (int)(r0 / (AA * NNB));     // same b for all 4 rows
            a0 = X0 + ((long)(b * AA + NI[r0])) * NFD + khalf;
            a1 = X0 + ((long)(b * AA + NI[r1])) * NFD + khalf;
            a2 = X0 + ((long)(b * AA + NI[r2])) * NFD + khalf;
            a3 = X0 + ((long)(b * AA + NI[r3])) * NFD + khalf;
            if (MODE == 2) {
                c0 = X1 + r0 * (long)NFD + khalf;
                c1 = X1 + r1 * (long)NFD + khalf;
                c2 = X1 + r2 * (long)NFD + khalf;
                c3 = X1 + r3 * (long)NFD + khalf;
            }
        }
    }

    v8f acc[4][2];
#pragma unroll
    for (int t = 0; t < 4; ++t)
#pragma unroll
        for (int i = 0; i < 8; ++i) { acc[t][0][i] = bv0; acc[t][1][i] = bv1; }

    // One K-step: 4 A b64 loads + 4 B b32 loads + 8 WMMAs.
    auto step = [&](gcf& A0, gcf& A1, gcf& A2, gcf& A3) {
        v2f av0, av1, av2, av3, b0, b1;
        av0[0] = A0[0]; av0[1] = A0[1]; A0 += 4;
        av1[0] = A1[0]; av1[1] = A1[1]; A1 += 4;
        av2[0] = A2[0]; av2[1] = A2[1]; A2 += 4;
        av3[0] = A3[0]; av3[1] = A3[1]; A3 += 4;
        b0[0] = bp0[0]; b0[1] = bp0[NC]; bp0 += 4 * NC;
        b1[0] = bp1[0]; b1[1] = bp1[NC]; bp1 += 4 * NC;
        acc[0][0] = __builtin_amdgcn_wmma_f32_16x16x4_f32(false, av0, false, b0, (short)0, acc[0][0], false, false);
        acc[0][1] = __builtin_amdgcn_wmma_f32_16x16x4_f32(false, av0, false, b1, (short)0, acc[0][1], false, false);
        acc[1][0] = __builtin_amdgcn_wmma_f32_16x16x4_f32(false, av1, false, b0, (short)0, acc[1][0], false, false);
        acc[1][1] = __builtin_amdgcn_wmma_f32_16x16x4_f32(false, av1, false, b1, (short)0, acc[1][1], false, false);
        acc[2][0] = __builtin_amdgcn_wmma_f32_16x16x4_f32(false, av2, false, b0, (short)0, acc[2][0], false, false);
        acc[2][1] = __builtin_amdgcn_wmma_f32_16x16x4_f32(false, av2, false, b1, (short)0, acc[2][1], false, false);
        acc[3][0] = __builtin_amdgcn_wmma_f32_16x16x4_f32(false, av3, false, b0, (short)0, acc[3][0], false, false);
        acc[3][1] = __builtin_amdgcn_wmma_f32_16x16x4_f32(false, av3, false, b1, (short)0, acc[3][1], false, false);
    };

    constexpr int LEN0 = (MODE == 2) ? NFD : K;
#pragma unroll 4
    for (int kk = 0; kk < LEN0; kk += 4) step(a0, a1, a2, a3);
    if constexpr (MODE == 2) {
#pragma unroll 4
        for (int kk = 0; kk < K - NFD; kk += 4) step(c0, c1, c2, c3);
    }

    // ---- activation + store (D: lane<16 -> M=i, lane>=16 -> M=8+i; N=nl) ----
#pragma unroll
    for (int t = 0; t < 4; ++t) {
        const long orow0 = mB + t * 16 + ((lane >> 4) << 3);
#pragma unroll
        for (int u = 0; u < 2; ++u) {
            v8f c = acc[t][u];
            if (ACT == 1) {
#pragma unroll
                for (int i = 0; i < 8; ++i) c[i] = fmaxf(c[i], 0.0f);
            } else if (ACT == 2) {
#pragma unroll
                for (int i = 0; i < 8; ++i) c[i] = c[i] / (1.0f + expf(-c[i]));
            }
            const int col = (u == 0) ? col0 : col1;
#pragma unroll
            for (int i = 0; i < 8; ++i)
                OUT[(orow0 + i) * (long)NC + col] = c[i];
        }
    }
}

template<int MODE, int K, int NC, int ACT>
static void run_gemm(const float* X0, const float* X1, const int* Ni,
                     const float* W, const float* bias, float* O,
                     long R, hipStream_t s)
{
    const long tiles = (R / 64) * (NC / 32);
    const unsigned blocks = (unsigned)((tiles + 7) / 8);   // 8 waves / block
    wmma_mlp_gemm<MODE, K, NC, ACT><<<dim3(blocks), dim3(256), 0, s>>>(
        X0, X1, Ni, W, bias, O, (int)R);
}

// ---------------------------------------------------------------------------
// Cosine cutoff applied in-place to rbf_msij, then msij = rbf*aj*ai.
// grid = B*A*NN blocks, 128 threads.
// ---------------------------------------------------------------------------
__global__ void msij_kernel(float* __restrict__ rbf_msij,
                            const float* __restrict__ a_msij,
                            const float* __restrict__ dist,
                            const int*   __restrict__ Nidx,
                            float* __restrict__ msij)
{
    const long row = blockIdx.x;
    const int  f   = threadIdx.x;
    const int  b   = (int)(row / (AA * NNB));
    const int  ai  = (int)((row / NNB) % AA);
    const int  j   = togi(Nidx)[row];

    const float d  = togc(dist)[row];
    const float Cv = (d < CUT_R) ? 0.5f * (cosf(d * PI_F / CUT_R) + 1.0f) : 0.0f;

    const long ro = row * NFD + f;
    const float rb = togc(rbf_msij)[ro] * Cv;
    tog(rbf_msij)[ro] = rb;
    const float ajm = togc(a_msij)[((long)(b * AA + j)) * NFD + f];
    const float aim = togc(a_msij)[((long)(b * AA + ai)) * NFD + f];
    tog(msij)[ro] = rb * ajm * aim;
}

// ---------------------------------------------------------------------------
// Non-linear attention per (b,a): att = softmax_n(sum_head(q*k) + maskinf),
// out = a + sum_n att*v. grid = B*A blocks, 256 threads.
// ---------------------------------------------------------------------------
__global__ void attention_kernel(const float* __restrict__ a,
                                 const float* __restrict__ q,
                                 const float* __restrict__ kbuf,
                                 const float* __restrict__ vbuf,
                                 const int*   __restrict__ NM,
                                 float* __restrict__ a_att)
{
    const int ba = blockIdx.x;
    const int t  = threadIdx.x;
    __shared__ float qs[NFD];
    __shared__ float att[NNB * NHEAD];

    if (t < NFD) qs[t] = togc(q)[(long)ba * NFD + t];
    __syncthreads();

    {   // att[n][h]
        const int n = t >> 2;
        const int h = t & 3;
        gcf krow = togc(kbuf) + ((long)ba * NNB + n) * NFD + h * (NFD / NHEAD);
        float s = 0.0f;
#pragma unroll
        for (int fo = 0; fo < NFD / NHEAD; ++fo)
            s += qs[h * (NFD / NHEAD) + fo] * krow[fo];
        if (togi(NM)[(long)ba * NNB + n] == 0) s += -100000.0f;
        att[n * NHEAD + h] = s;
    }
    __syncthreads();

    if (t < NHEAD) {   // softmax over neighbors for head t
        float mx = -INFINITY;
        for (int n = 0; n < NNB; ++n) mx = fmaxf(mx, att[n * NHEAD + t]);
        float sum = 0.0f;
        for (int n = 0; n < NNB; ++n) {
            const float e = expf(att[n * NHEAD + t] - mx);
            att[n * NHEAD + t] = e;
            sum += e;
        }
        const float inv = 1.0f / sum;
        for (int n = 0; n < NNB; ++n) att[n * NHEAD + t] *= inv;
    }
    __syncthreads();

    if (t < NFD) {
        const int h = t / (NFD / NHEAD);
        gcf vcol = togc(vbuf) + (long)ba * NNB * NFD + t;
        float acc = 0.0f;
        for (int n = 0; n < NNB; ++n)
            acc += att[n * NHEAD + h] * vcol[(long)n * NFD];
        tog(a_att)[(long)ba * NFD + t] = togc(a)[(long)ba * NFD + t] + acc;
    }
}

// ---------------------------------------------------------------------------
// F_ij = (msij . f_w) * distance_vector ; f_dir_out = f_dir_in + sum_n NM*F_ij
// grid = B*A blocks, 64 threads (one per neighbor)
// ---------------------------------------------------------------------------
__global__ void fij_kernel(const float* __restrict__ msij,
                           const float* __restrict__ f_w,
                           const float* __restrict__ dv,
                           const int*   __restrict__ NM,
                           const float* __restrict__ f_dir_in,
                           float* __restrict__ F_ij,
                           float* __restrict__ out_fdir)
{
    const int  ba  = blockIdx.x;
    const int  n   = threadIdx.x;
    const long row = (long)ba * NNB + n;

    float s = 0.0f;
    gcf mrow = togc(msij) + row * NFD;
#pragma unroll 8
    for (int f = 0; f < NFD; ++f) s += mrow[f] * togc(f_w)[f];

    float fv[3];
#pragma unroll
    for (int d = 0; d < 3; ++d) {
        fv[d] = s * togc(dv)[row * 3 + d];
        tog(F_ij)[row * 3 + d] = fv[d];
    }
    const float m = (float)togi(NM)[row];

    __shared__ float red[3][NNB];
#pragma unroll
    for (int d = 0; d < 3; ++d) red[d][n] = m * fv[d];
    __syncthreads();
    if (n < 3) {
        float acc = 0.0f;
        for (int i = 0; i < NNB; ++i) acc += red[n][i];
        tog(out_fdir)[(long)ba * 3 + n] = togc(f_dir_in)[(long)ba * 3 + n] + acc;
    }
}

// ---------------------------------------------------------------------------
// F_i[ba,d,f] = sum_n NM * fscale[ba,n,f] * F_ij[ba,n,d]
// grid = B*A blocks, 128 threads (one per f)
// ---------------------------------------------------------------------------
__global__ void fi_kernel(const float* __restrict__ fscale,
                          const float* __restrict__ F_ij,
                          const int*   __restrict__ NM,
                          float* __restrict__ F_i)
{
    const int ba = blockIdx.x;
    const int f  = threadIdx.x;
    float a0 = 0.0f, a1 = 0.0f, a2 = 0.0f;
    for (int n = 0; n < NNB; ++n) {
        const long row = (long)ba * NNB + n;
        const float fs = (float)togi(NM)[row] * togc(fscale)[row * NFD + f];
        a0 += fs * togc(F_ij)[row * 3 + 0];
        a1 += fs * togc(F_ij)[row * 3 + 1];
        a2 += fs * togc(F_ij)[row * 3 + 2];
    }
    tog(F_i)[((long)ba * 3 + 0) * NFD + f] = a0;
    tog(F_i)[((long)ba * 3 + 1) * NFD + f] = a1;
    tog(F_i)[((long)ba * 3 + 2) * NFD + f] = a2;
}

// ---------------------------------------------------------------------------
// dr_ext[ba,d,f] = sum_n NM * rext[ba,n,f] * r_dyn[b, N[ba,n], d, f]
// grid = B*A blocks, 128 threads
// ---------------------------------------------------------------------------
__global__ void drext_kernel(const float* __restrict__ rext,
                             const int*   __restrict__ Nidx,
                             const int*   __restrict__ NM,
                             const float* __restrict__ r_dyn_in,
                             float* __restrict__ dr_ext)
{
    const int ba = blockIdx.x;
    const int f  = threadIdx.x;
    const int b  = ba / AA;
    float a0 = 0.0f, a1 = 0.0f, a2 = 0.0f;
    for (int n = 0; n < NNB; ++n) {
        const long row = (long)ba * NNB + n;
        const float re = (float)togi(NM)[row] * togc(rext)[row * NFD + f];
        const long base = ((long)(b * AA + togi(Nidx)[row])) * 3 * NFD + f;
        a0 += re * togc(r_dyn_in)[base + 0 * NFD];
        a1 += re * togc(r_dyn_in)[base + 1 * NFD];
        a2 += re * togc(r_dyn_in)[base + 2 * NFD];
    }
    tog(dr_ext)[((long)ba * 3 + 0) * NFD + f] = a0;
    tog(dr_ext)[((long)ba * 3 + 1) * NFD + f] = a1;
    tog(dr_ext)[((long)ba * 3 + 2) * NFD + f] = a2;
}

// ---------------------------------------------------------------------------
// Final dynamics combine + outputs. grid = B*A blocks, 128 threads.
// ---------------------------------------------------------------------------
__global__ void final_kernel(const float* __restrict__ a_att,
                             const float* __restrict__ F_i,
                             const float* __restrict__ dr_ext,
                             const float* __restrict__ rgate,
                             const float* __restrict__ egate,
                             const float* __restrict__ f_dyn_in,
                             const float* __restrict__ r_dyn_in,
                             const float* __restrict__ e_dyn_in,
                             float* __restrict__ out_a,
                             float* __restrict__ out_fdyn,
                             float* __restrict__ out_rdyn,
                             float* __restrict__ out_edyn)
{
    const int  ba = blockIdx.x;
    const int  f  = threadIdx.x;
    const float rg = togc(rgate)[(long)ba * NFD + f];

    float dot = 0.0f;
#pragma unroll
    for (int d = 0; d < 3; ++d) {
        const long idx = ((long)ba * 3 + d) * NFD + f;
        const float Fi = togc(F_i)[idx];
        const float fd = togc(f_dyn_in)[idx] + Fi;
        const float rd = togc(r_dyn_in)[idx] + rg * Fi + togc(dr_ext)[idx];
        tog(out_fdyn)[idx] = fd;
        tog(out_rdyn)[idx] = rd;
        dot += fd * rd;
    }
    const float de = -dot * togc(egate)[(long)ba * NFD + f];
    tog(out_a)[(long)ba * NFD + f]    = togc(a_att)[(long)ba * NFD + f] + de;
    tog(out_edyn)[(long)ba * NFD + f] = togc(e_dyn_in)[(long)ba * NFD + f] + de;
}

// ---------------------------------------------------------------------------
extern "C" void kernel_launch(void* const* d_in, const int* in_sizes, int n_in,
                              void* d_out, int out_size, void* d_ws, size_t ws_size,
                              hipStream_t stream)
{
    // ---- inputs (setup_inputs dict order, params flattened in dict order) ----
    const float* a_in     = (const float*)d_in[0];
    const float* rbf      = (const float*)d_in[1];
    const float* dist     = (const float*)d_in[2];
    const float* dvec     = (const float*)d_in[3];
    const int*   Nidx     = (const int*)d_in[4];
    const int*   NM       = (const int*)d_in[5];
    const float* fdir_in  = (const float*)d_in[6];
    const float* fdyn_in  = (const float*)d_in[7];
    const float* rdyn_in  = (const float*)d_in[8];
    const float* edyn_in  = (const float*)d_in[9];
    const float* rbf_w = (const float*)d_in[10]; const float* rbf_b = (const float*)d_in[11];
    const float* a1_w  = (const float*)d_in[12]; const float* a1_b  = (const float*)d_in[13];
    const float* a2_w  = (const float*)d_in[14]; const float* a2_b  = (const float*)d_in[15];
    const float* q1_w  = (const float*)d_in[16]; const float* q1_b  = (const float*)d_in[17];
    const float* q2_w  = (const float*)d_in[18]; const float* q2_b  = (const float*)d_in[19];
    const float* k1_w  = (const float*)d_in[20]; const float* k1_b  = (const float*)d_in[21];
    const float* k2_w  = (const float*)d_in[22]; const float* k2_b  = (const float*)d_in[23];
    const float* v1_w  = (const float*)d_in[24]; const float* v1_b  = (const float*)d_in[25];
    const float* v2_w  = (const float*)d_in[26]; const float* v2_b  = (const float*)d_in[27];
    const float* f_w   = (const float*)d_in[28];
    const float* fs1_w = (const float*)d_in[29]; const float* fs1_b = (const float*)d_in[30];
    const float* fs2_w = (const float*)d_in[31]; const float* fs2_b = (const float*)d_in[32];
    const float* r1_w  = (const float*)d_in[33]; const float* r1_b  = (const float*)d_in[34];
    const float* r2_w  = (const float*)d_in[35]; const float* r2_b  = (const float*)d_in[36];
    const float* re1_w = (const float*)d_in[37]; const float* re2_w = (const float*)d_in[38];
    const float* e1_w  = (const float*)d_in[39]; const float* e1_b  = (const float*)d_in[40];
    const float* e2_w  = (const float*)d_in[41]; const float* e2_b  = (const float*)d_in[42];

    // ---- workspace layout (floats) ----
    float* w = (float*)d_ws;
    const long R_NB  = (long)BB * AA * NNB;          // 131072
    const long R_AT  = (long)BB * AA;                // 2048
    long off = 0;
    float* RBF    = w + off; off += R_NB * NFD;      // rbf_msij (cutoff in place)
    float* MSIJ   = w + off; off += R_NB * NFD;
    float* H1     = w + off; off += R_NB * 256;      // big hidden scratch
    float* KOUT   = w + off; off += R_NB * NFD;      // k, later fscale
    float* VOUT   = w + off; off += R_NB * NFD;      // v, later rext
    float* AMSIJ  = w + off; off += R_AT * NFD;
    float* QBUF   = w + off; off += R_AT * NFD;
    float* AATT   = w + off; off += R_AT * NFD;
    float* RGATE  = w + off; off += R_AT * NFD;
    float* EGATE  = w + off; off += R_AT * NFD;
    float* HS     = w + off; off += R_AT * 256;      // small per-atom hidden
    float* FIJ    = w + off; off += R_NB * 3;
    float* FI     = w + off; off += R_AT * 3 * NFD;
    float* DREXT  = w + off; off += R_AT * 3 * NFD;

    // ---- output layout (tuple concat, flat) ----
    float* out = (float*)d_out;
    float* out_a    = out;                                     // (B,A,NF)
    float* out_fdir = out_a    + R_AT * NFD;                   // (B,A,3)
    float* out_fdyn = out_fdir + R_AT * 3;                     // (B,A,3,NF)
    float* out_rdyn = out_fdyn + R_AT * 3 * NFD;               // (B,A,3,NF)
    float* out_edyn = out_rdyn + R_AT * 3 * NFD;               // (B,A,NF)

    // 1) rbf projection (raw, cutoff applied later)
    run_gemm<0, RESD, NFD, 0>(rbf, nullptr, nullptr, rbf_w, rbf_b, RBF, R_NB, stream);

    // 2) a_msij = silu(a@a1+b1)@a2+b2
    run_gemm<0, NFD, NFD, 2>(a_in, nullptr, nullptr, a1_w, a1_b, HS, R_AT, stream);
    run_gemm<0, NFD, NFD, 0>(HS,   nullptr, nullptr, a2_w, a2_b, AMSIJ, R_AT, stream);

    // 3) cutoff in place + msij = rbf*aj*ai
    msij_kernel<<<dim3((unsigned)R_NB), dim3(NFD), 0, stream>>>(RBF, AMSIJ, dist, Nidx, MSIJ);

    // 4) q = relu(a@q1+b1)@q2+b2      (original a)
    run_gemm<0, NFD, 256, 1>(a_in, nullptr, nullptr, q1_w, q1_b, HS, R_AT, stream);
    run_gemm<0, 256, NFD, 0>(HS,   nullptr, nullptr, q2_w, q2_b, QBUF, R_AT, stream);

    // 5) k = relu(aj_msij@k1+b1)@k2+b2   (gathered rows, MODE 1)
    run_gemm<1, NFD, 256, 1>(AMSIJ, nullptr, Nidx, k1_w, k1_b, H1, R_NB, stream);
    run_gemm<0, 256, NFD, 0>(H1,    nullptr, nullptr, k2_w, k2_b, KOUT, R_NB, stream);

    // 6) v = relu(concat(aj_msij, rbf_msij)@v1+b1)@v2+b2   (MODE 2)
    run_gemm<2, 256, 256, 1>(AMSIJ, RBF, Nidx, v1_w, v1_b, H1, R_NB, stream);
    run_gemm<0, 256, NFD, 0>(H1,    nullptr, nullptr, v2_w, v2_b, VOUT, R_NB, stream);

    // 7) attention -> a_att
    attention_kernel<<<dim3((unsigned)R_AT), dim3(256), 0, stream>>>(
        a_in, QBUF, KOUT, VOUT, NM, AATT);

    // 8) gates on post-attention a
    run_gemm<0, NFD, NFD, 2>(AATT, nullptr, nullptr, r1_w, r1_b, HS, R_AT, stream);
    run_gemm<0, NFD, NFD, 0>(HS,   nullptr, nullptr, r2_w, r2_b, RGATE, R_AT, stream);
    run_gemm<0, NFD, NFD, 2>(AATT, nullptr, nullptr, e1_w, e1_b, HS, R_AT, stream);
    run_gemm<0, NFD, NFD, 0>(HS,   nullptr, nullptr, e2_w, e2_b, EGATE, R_AT, stream);

    // 9) F_ij and f_dir output
    fij_kernel<<<dim3((unsigned)R_AT), dim3(NNB), 0, stream>>>(
        MSIJ, f_w, dvec, NM, fdir_in, FIJ, out_fdir);

    // 10) fscale (reuses H1 + KOUT)
    run_gemm<0, NFD, NFD, 2>(MSIJ, nullptr, nullptr, fs1_w, fs1_b, H1, R_NB, stream);
    run_gemm<0, NFD, NFD, 0>(H1,   nullptr, nullptr, fs2_w, fs2_b, KOUT, R_NB, stream);

    // 11) rext = silu(msij@re1)@re2, no biases (reuses H1 + VOUT)
    run_gemm<0, NFD, NFD, 2>(MSIJ, nullptr, nullptr, re1_w, nullptr, H1, R_NB, stream);
    run_gemm<0, NFD, NFD, 0>(H1,   nullptr, nullptr, re2_w, nullptr, VOUT, R_NB, stream);

    // 12) neighbor reductions
    fi_kernel<<<dim3((unsigned)R_AT), dim3(NFD), 0, stream>>>(KOUT, FIJ, NM, FI);
    drext_kernel<<<dim3((unsigned)R_AT), dim3(NFD), 0, stream>>>(VOUT, Nidx, NM, rdyn_in, DREXT);

    // 13) final combine + remaining outputs
    final_kernel<<<dim3((unsigned)R_AT), dim3(NFD), 0, stream>>>(
        AATT, FI, DREXT, RGATE, EGATE, fdyn_in, rdyn_in, edyn_in,
        out_a, out_fdyn, out_rdyn, out_edyn);

    (void)in_sizes; (void)n_in; (void)out_size; (void)ws_size;
}